// EuclideanCodebook_19997367730537
// MI455X (gfx1250) — compile-verified
//
#include <hip/hip_runtime.h>

typedef __attribute__((ext_vector_type(2))) float v2f;
typedef __attribute__((ext_vector_type(4))) float v4f;
typedef __attribute__((ext_vector_type(8))) float v8f;
typedef unsigned int u32x4 __attribute__((ext_vector_type(4)));
typedef int i32x8 __attribute__((ext_vector_type(8)));
typedef int i32x4 __attribute__((ext_vector_type(4)));

#define DIM 128
#define KCODES 1024
#define WAVES_PER_BLOCK 8
#define ROWS_PER_WAVE 16
#define TILE_CODES 64                      // codes staged per LDS tile
#define N_STAGES (KCODES / TILE_CODES)     // 16
#define ROW_PITCH 132                      // 128 dwords + 4 dwords TDM pad (bank-conflict-free)
#define TILE_FLOATS (TILE_CODES * ROW_PITCH)
#define TILE_BYTES (TILE_FLOATS * 4)

#if defined(__has_builtin)
#if __has_builtin(__builtin_amdgcn_tensor_load_to_lds)
#define HAVE_TDM 1
#endif
#endif

// ---------------------------------------------------------------------------
// Kernel 1: e2[k] = 0.5 * ||embed[k]||^2   (1024 floats into d_ws)
// ---------------------------------------------------------------------------
__global__ __launch_bounds__(256) void vq_e2_kernel(const float* __restrict__ embed,
                                                    float* __restrict__ e2) {
    int code = blockIdx.x * blockDim.x + threadIdx.x;
    if (code >= KCODES) return;
    const v4f* p = (const v4f*)(embed + (size_t)code * DIM);
    float s = 0.0f;
#pragma unroll
    for (int j = 0; j < DIM / 4; ++j) {
        v4f t = p[j];
        s += t[0] * t[0] + t[1] * t[1] + t[2] * t[2] + t[3] * t[3];
    }
    e2[code] = 0.5f * s;
}

#ifdef HAVE_TDM
// ---------------------------------------------------------------------------
// Issue one TDM 2D tile load: embed rows [crow, crow+64) x 128 f32 -> LDS,
// with 4-dword padding after every 128 dwords (LDS row pitch = 132 dwords).
// All descriptor fields are wave-uniform (land in SGPRs). Tracked by TENSORcnt.
// 6-arg builtin form (clang-23 / therock-10.0 lane).
// ---------------------------------------------------------------------------
__device__ __forceinline__ void tdm_stage(const float* gsrc, unsigned lds_byte_off) {
    unsigned long long ga = (unsigned long long)(size_t)gsrc;
    u32x4 g0;
    g0[0] = 1u;                                        // count=1 (valid), user mode
    g0[1] = lds_byte_off;                              // lds_addr
    g0[2] = (unsigned)(ga & 0xFFFFFFFFu);              // global_addr[31:0]
    g0[3] = (unsigned)((ga >> 32) & 0x01FFFFFFu)       // global_addr[56:32]
            | (2u << 30);                              // type=2 ("image")
    i32x8 g1;
    g1[0] = (int)((2u << 16)      // data_size = 4 bytes
                | (1u << 20)      // pad_enable
                | (6u << 22)      // pad_interval: 2^(6+1)=128 dwords
                | (3u << 25));    // pad_amount: 3+1 = 4 dwords
    g1[1] = (int)(128u << 16);    // tensor_dim0 = 128 (low16 in [63:48])
    g1[2] = (int)(1024u << 16);   // tensor_dim1 = 1024 (low16 in [95:80])
    g1[3] = (int)(128u << 16);    // tile_dim0 = 128
    g1[4] = TILE_CODES;           // tile_dim1 = 64, tile_dim2 = 0
    g1[5] = 128;                  // tensor_dim0_stride = 128 (low32)
    g1[6] = 0;
    g1[7] = 0;
    i32x4 gz4 = {0, 0, 0, 0};                 // groups 2/3: unused (2D tile)
    i32x8 gz8 = {0, 0, 0, 0, 0, 0, 0, 0};     // extra group: unused
    __builtin_amdgcn_tensor_load_to_lds(g0, g1, gz4, gz4, gz8, 0);
}
#endif

// ---------------------------------------------------------------------------
// Per-stage compute: 4 independent 32-deep WMMA chains (codes cbase..cbase+63),
// B fragments from LDS (ds_load_b64, conflict-free with pitch 132).
// ---------------------------------------------------------------------------
__device__ __forceinline__ void compute_tile(const float* __restrict__ buf,
                                             const v2f* __restrict__ a,
                                             int cbase,
                                             const float* __restrict__ e2,
                                             float* __restrict__ best_val,
                                             int* __restrict__ best_idx,
                                             int l15, int half) {
    const float* b0 = buf + (l15 +  0) * ROW_PITCH + 2 * half;
    const float* b1 = buf + (l15 + 16) * ROW_PITCH + 2 * half;
    const float* b2 = buf + (l15 + 32) * ROW_PITCH + 2 * half;
    const float* b3 = buf + (l15 + 48) * ROW_PITCH + 2 * half;
    v8f acc0 = {}, acc1 = {}, acc2 = {}, acc3 = {};
#pragma unroll
    for (int j = 0; j < 32; ++j) {
        v2f aj = a[j];
        acc0 = __builtin_amdgcn_wmma_f32_16x16x4_f32(
            false, aj, false, *(const v2f*)(b0 + 4 * j), (short)0, acc0, false, false);
        acc1 = __builtin_amdgcn_wmma_f32_16x16x4_f32(
            false, aj, false, *(const v2f*)(b1 + 4 * j), (short)0, acc1, false, false);
        acc2 = __builtin_amdgcn_wmma_f32_16x16x4_f32(
            false, aj, false, *(const v2f*)(b2 + 4 * j), (short)0, acc2, false, false);
        acc3 = __builtin_amdgcn_wmma_f32_16x16x4_f32(
            false, aj, false, *(const v2f*)(b3 + 4 * j), (short)0, acc3, false, false);
    }
    const int c0 = cbase + l15, c1 = c0 + 16, c2 = c0 + 32, c3 = c0 + 48;
    const float h0 = e2[c0], h1 = e2[c1], h2 = e2[c2], h3 = e2[c3];
#pragma unroll
    for (int r = 0; r < 8; ++r) {
        float s0 = acc0[r] - h0;
        float s1 = acc1[r] - h1;
        float s2 = acc2[r] - h2;
        float s3 = acc3[r] - h3;
        if (s0 > best_val[r]) { best_val[r] = s0; best_idx[r] = c0; }
        if (s1 > best_val[r]) { best_val[r] = s1; best_idx[r] = c1; }
        if (s2 > best_val[r]) { best_val[r] = s2; best_idx[r] = c2; }
        if (s3 > best_val[r]) { best_val[r] = s3; best_idx[r] = c3; }
    }
}

// ---------------------------------------------------------------------------
// Kernel 2: fused GEMM (f32 WMMA 16x16x4) + row argmax + gather.
// Block = 8 waves x 16 rows; embed staged per-block into LDS by the TDM
// (double-buffered, overlapped with compute).
//   score(row, code) = x.e - 0.5*||e||^2   (||x||^2 row-constant, dropped)
// ---------------------------------------------------------------------------
__global__ __launch_bounds__(256) void vq_argmax_kernel(const float* __restrict__ x,
                                                        const float* __restrict__ embed,
                                                        const float* __restrict__ e2,
                                                        float* __restrict__ quant,
                                                        float* __restrict__ ind,
                                                        int n_rows) {
    extern __shared__ float lds_tiles[];   // 2 * TILE_FLOATS

    const int lane = threadIdx.x & 31;
    const int wave = threadIdx.x >> 5;
    const int row_base = (blockIdx.x * WAVES_PER_BLOCK + wave) * ROWS_PER_WAVE;
    if (row_base >= n_rows) return;

    const int half = lane >> 4;   // which K-pair of the 16x16x4 step this lane feeds
    const int l15  = lane & 15;

    // Preload A: 16 rows x 128 K -> 32 x float2 per lane (64 VGPRs), per ISA
    // 16x4 fp32 A layout (lane<16: K={4j,4j+1}; lane>=16: K={4j+2,4j+3}).
    const float* xrow = x + (size_t)(row_base + l15) * DIM + 2 * half;
    v2f a[32];
#pragma unroll
    for (int j = 0; j < 32; ++j)
        a[j] = *(const v2f*)(xrow + 4 * j);

    float best_val[8];
    int   best_idx[8];
#pragma unroll
    for (int r = 0; r < 8; ++r) { best_val[r] = -3.4e38f; best_idx[r] = 0; }

#ifdef HAVE_TDM
    const unsigned lds_base = __builtin_amdgcn_groupstaticsize();  // dynamic LDS start
    if (wave == 0) {
        tdm_stage(embed, lds_base);                    // stage 0 -> buffer 0
        __builtin_amdgcn_s_wait_tensorcnt(0);
    }
    __syncthreads();
    for (int s = 0; s < N_STAGES; ++s) {
        if (wave == 0 && s + 1 < N_STAGES)             // prefetch next tile (async)
            tdm_stage(embed + (size_t)(s + 1) * TILE_CODES * DIM,
                      lds_base + (unsigned)(((s + 1) & 1) * TILE_BYTES));
        compute_tile(lds_tiles + (s & 1) * TILE_FLOATS, a, s * TILE_CODES,
                     e2, best_val, best_idx, l15, half);
        if (wave == 0)
            __builtin_amdgcn_s_wait_tensorcnt(0);      // next buffer resident
        __syncthreads();                               // all waves done with buf s&1
    }
#else
    // Fallback: synchronous cooperative staging (still LDS-shared across waves).
    for (int s = 0; s < N_STAGES; ++s) {
        __syncthreads();
        const float* src = embed + (size_t)s * TILE_CODES * DIM;
        for (int k = threadIdx.x; k < TILE_CODES * DIM; k += 256) {
            int r = k >> 7, c = k & 127;
            lds_tiles[r * ROW_PITCH + c] = src[k];
        }
        __syncthreads();
        compute_tile(lds_tiles, a, s * TILE_CODES, e2, best_val, best_idx, l15, half);
    }
#endif

    // Argmax reduction across the 16 lanes of each half (rows 0-7 / 8-15).
    // First-index tie-break matches jnp.argmax.
#pragma unroll
    for (int r = 0; r < 8; ++r) {
        float v = best_val[r];
        int   i = best_idx[r];
#pragma unroll
        for (int m = 1; m < 16; m <<= 1) {
            float ov = __shfl_xor(v, m, 16);
            int   oi = __shfl_xor(i, m, 16);
            if (ov > v || (ov == v && oi < i)) { v = ov; i = oi; }
        }
        best_idx[r] = i;
    }

    // Indices (as float: d_out is one flat float buffer holding both outputs).
    if (l15 == 0) {
#pragma unroll
        for (int r = 0; r < 8; ++r)
            ind[row_base + half * 8 + r] = (float)best_idx[r];
    }

    // Broadcast the 16 winning code ids to the wave, then gather quantize rows
    // cooperatively: 128 floats = 32 lanes x float4 (b128 loads/stores).
    int rowidx[16];
#pragma unroll
    for (int r = 0; r < 8; ++r) {
        rowidx[r]     = __shfl(best_idx[r], 0, 32);
        rowidx[r + 8] = __shfl(best_idx[r], 16, 32);
    }
#pragma unroll
    for (int rr = 0; rr < 16; ++rr) {
        const v4f* src = (const v4f*)(embed + (size_t)rowidx[rr] * DIM) + lane;
        v4f*       dst = (v4f*)(quant + (size_t)(row_base + rr) * DIM) + lane;
        *dst = *src;
    }
}

// ---------------------------------------------------------------------------
extern "C" void kernel_launch(void* const* d_in, const int* in_sizes, int n_in,
                              void* d_out, int out_size, void* d_ws, size_t ws_size,
                              hipStream_t stream) {
    const float* x     = (const float*)d_in[0];   // [B*T, 128]
    const float* embed = (const float*)d_in[1];   // [1024, 128]
    const int n_rows = in_sizes[0] / DIM;         // 96000

    float* quant = (float*)d_out;                           // [n_rows, 128]
    float* ind   = (float*)d_out + (size_t)n_rows * DIM;    // [n_rows]
    float* e2    = (float*)d_ws;                            // [1024]

    vq_e2_kernel<<<(KCODES + 255) / 256, 256, 0, stream>>>(embed, e2);

    const int n_rg     = (n_rows + ROWS_PER_WAVE - 1) / ROWS_PER_WAVE;
    const int n_blocks = (n_rg + WAVES_PER_BLOCK - 1) / WAVES_PER_BLOCK;
    const size_t lds_bytes = 2u * TILE_BYTES;               // double-buffered tiles
    vq_argmax_kernel<<<n_blocks, 256, lds_bytes, stream>>>(x, embed, e2, quant, ind, n_rows);
}